// VFR_61538291417252
// MI455X (gfx1250) — compile-verified
//
#include <hip/hip_runtime.h>
#include <math.h>

typedef __attribute__((ext_vector_type(2))) float v2f;
typedef __attribute__((ext_vector_type(8))) float v8f;

#define BATCH   8
#define NPTS    4096
#define KNN     32
#define CH      64
#define ROWS    (BATCH * NPTS)          // 32768
#define GM_BLOCKS 512                   // gather_max grid (64 points/block)

// ---------------------------------------------------------------------------
// Kernel A: h[g, :] = x[g, :] @ W^T + bias   via V_WMMA_F32_16X16X4_F32
// One wave per 16-row tile. A frags resident (16 ksteps x v2f), loop 4 N-tiles.
// ---------------------------------------------------------------------------
__global__ void __launch_bounds__(256)
linear_wmma(const float* __restrict__ x, const float* __restrict__ W,
            const float* __restrict__ bias, float* __restrict__ h) {
  const int lane = threadIdx.x & 31;
  const int wave = threadIdx.x >> 5;
  const int tile = blockIdx.x * 8 + wave;      // 0..2047, no remainder
  const int m0   = tile * 16;
  const int half = lane >> 4;                  // 0: K+0/K+1, 1: K+2/K+3
  const int l15  = lane & 15;
  const int koff = half * 2;

  // A fragments: row = m0 + l15 for all lanes; kstep s covers K=4s..4s+3
  v2f a[16];
  const float* xrow = x + (size_t)(m0 + l15) * CH;
#pragma unroll
  for (int s = 0; s < 16; ++s)
    a[s] = *(const v2f*)(xrow + 4 * s + koff);

#pragma unroll
  for (int nt = 0; nt < 4; ++nt) {
    const int n0 = nt * 16;
    // B frag: lane holds W[n0+l15][k0+koff], W[n0+l15][k0+koff+1]
    const float* wrow = W + (size_t)(n0 + l15) * CH + koff;
    v8f c = {0.f, 0.f, 0.f, 0.f, 0.f, 0.f, 0.f, 0.f};
#pragma unroll
    for (int s = 0; s < 16; ++s) {
      v2f bf = *(const v2f*)(wrow + 4 * s);
      c = __builtin_amdgcn_wmma_f32_16x16x4_f32(
          false, a[s], false, bf, (short)0, c, false, false);
    }
    const float bv = bias[n0 + l15];
    // C/D layout: VGPR i -> M = i + 8*half, N = n0 + l15
    float* hout = h + (size_t)(m0 + half * 8) * CH + n0 + l15;
#pragma unroll
    for (int i = 0; i < 8; ++i)
      hout[(size_t)i * CH] = c[i] + bv;
  }
}

// ---------------------------------------------------------------------------
// Kernel B: agg[g,c] = max_k h[b, idx[g,k], c] - h[g,c]  (max distributes over
// the constant center subtraction), plus per-block channel sum / sumsq.
// One wave per point: lane l owns channels 2l, 2l+1 (float2), so each neighbor
// row is a single 256B contiguous b64-per-lane fetch. knn indices loaded once
// per point (lane == k) and broadcast with __shfl.
// ---------------------------------------------------------------------------
__global__ void __launch_bounds__(256)
gather_max(const float* __restrict__ h, const int* __restrict__ knn,
           float* __restrict__ agg, float* __restrict__ partial) {
  __shared__ float red[1024];        // [sum|sq][wave 0..7][channel 0..63]
  const int tid  = threadIdx.x;
  const int lane = tid & 31;
  const int wave = tid >> 5;         // 0..7
  const int c0   = 2 * lane;

  float2 lsum = make_float2(0.f, 0.f);
  float2 lsq  = make_float2(0.f, 0.f);

#pragma unroll 1
  for (int it = 0; it < 8; ++it) {                  // 8 points per wave
    const int g = blockIdx.x * 64 + it * 8 + wave;  // global point 0..32767
    const int b = g >> 12;                          // g / NPTS
    const float* hb = h + (size_t)b * NPTS * CH;
    const float2 center = *(const float2*)(h + (size_t)g * CH + c0);
    const int myidx = knn[(size_t)g * KNN + lane];  // lane == its k
    float2 m = make_float2(-INFINITY, -INFINITY);
#pragma unroll
    for (int k = 0; k < KNN; ++k) {
      const int idx = __shfl(myidx, k, 32);
      const float2 nv = *(const float2*)(hb + (size_t)idx * CH + c0);
      m.x = fmaxf(m.x, nv.x);
      m.y = fmaxf(m.y, nv.y);
    }
    const float2 v = make_float2(m.x - center.x, m.y - center.y);
    *(float2*)(agg + (size_t)g * CH + c0) = v;
    lsum.x += v.x;       lsum.y += v.y;
    lsq.x  += v.x * v.x; lsq.y  += v.y * v.y;
  }
  // per-wave channel partials -> LDS (stride-2 writes: 32 distinct even banks)
  *(float2*)(red + wave * 64 + c0)       = lsum;
  *(float2*)(red + 512 + wave * 64 + c0) = lsq;
  __syncthreads();
  if (tid < 64) {
    float S = 0.f, Q = 0.f;
#pragma unroll
    for (int w = 0; w < 8; ++w) {
      S += red[w * 64 + tid];
      Q += red[512 + w * 64 + tid];
    }
    partial[(size_t)blockIdx.x * 128 + tid]      = S;
    partial[(size_t)blockIdx.x * 128 + 64 + tid] = Q;
  }
}

// ---------------------------------------------------------------------------
// Kernel C: fold 512 partials -> fused BN scale/shift per channel.
// stats[c] = gamma[c]*rstd ; stats[64+c] = beta[c] - mu*gamma[c]*rstd
// ---------------------------------------------------------------------------
__global__ void __launch_bounds__(256)
bn_stats(const float* __restrict__ partial, const float* __restrict__ gamma,
         const float* __restrict__ beta, float* __restrict__ stats) {
  __shared__ float red[512];
  const int tid = threadIdx.x;
  const int c   = tid & 63;
  const int grp = tid >> 6;  // 4 groups
  float s = 0.f, q = 0.f;
  for (int blk = grp; blk < GM_BLOCKS; blk += 4) {
    s += partial[(size_t)blk * 128 + c];
    q += partial[(size_t)blk * 128 + 64 + c];
  }
  red[tid]       = s;
  red[256 + tid] = q;
  __syncthreads();
  if (grp == 0) {
    const float S = red[c] + red[64 + c] + red[128 + c] + red[192 + c];
    const float Q = red[256 + c] + red[320 + c] + red[384 + c] + red[448 + c];
    const float inv_n = 1.0f / (float)ROWS;
    const float mu  = S * inv_n;
    const float var = Q * inv_n - mu * mu;     // biased variance
    const float rstd = rsqrtf(var + 1e-5f);
    const float scale = gamma[c] * rstd;
    stats[c]      = scale;
    stats[64 + c] = beta[c] - mu * scale;
  }
}

// ---------------------------------------------------------------------------
// Kernel D: in-place affine on d_out (which holds agg), float4 vectorized.
// CH=64 is divisible by 4, so a float4 never crosses a row boundary.
// ---------------------------------------------------------------------------
__global__ void __launch_bounds__(256)
bn_apply(float* __restrict__ out, const float* __restrict__ stats) {
  const int i  = blockIdx.x * blockDim.x + threadIdx.x;  // group of 4 floats
  const int c4 = i & 15;                                 // (i*4 & 63) / 4
  float4 v  = ((float4*)out)[i];
  const float4 sc = ((const float4*)stats)[c4];
  const float4 sh = ((const float4*)(stats + 64))[c4];
  v.x = v.x * sc.x + sh.x;
  v.y = v.y * sc.y + sh.y;
  v.z = v.z * sc.z + sh.z;
  v.w = v.w * sc.w + sh.w;
  ((float4*)out)[i] = v;
}

// ---------------------------------------------------------------------------
extern "C" void kernel_launch(void* const* d_in, const int* in_sizes, int n_in,
                              void* d_out, int out_size, void* d_ws, size_t ws_size,
                              hipStream_t stream) {
  const float* x     = (const float*)d_in[0];
  const int*   knn   = (const int*)d_in[1];
  const float* W     = (const float*)d_in[2];
  const float* bias  = (const float*)d_in[3];
  const float* gamma = (const float*)d_in[4];
  const float* beta  = (const float*)d_in[5];
  float* out = (float*)d_out;

  // Workspace layout: h (8 MB) | partials (256 KB) | stats (512 B)
  float* h       = (float*)d_ws;                        // ROWS*CH floats
  float* partial = h + (size_t)ROWS * CH;               // GM_BLOCKS*128 floats
  float* stats   = partial + (size_t)GM_BLOCKS * 128;   // 128 floats

  // A: 32768 rows / 16 rows-per-wave / 8 waves-per-block = 256 blocks
  linear_wmma<<<dim3(256), dim3(256), 0, stream>>>(x, W, bias, h);
  // B: 512 blocks x 8 waves x 8 points; agg written directly into d_out
  gather_max<<<dim3(GM_BLOCKS), dim3(256), 0, stream>>>(h, knn, out, partial);
  // C: single-block deterministic reduction
  bn_stats<<<dim3(1), dim3(256), 0, stream>>>(partial, gamma, beta, stats);
  // D: 2M floats / 4 / 256 = 2048 blocks, in place on d_out
  bn_apply<<<dim3((ROWS * CH) / 4 / 256), dim3(256), 0, stream>>>(out, stats);
}